// FeaturePropagation_70325794505118
// MI455X (gfx1250) — compile-verified
//
#include <hip/hip_runtime.h>
#include <hip/hip_bf16.h>

typedef __attribute__((ext_vector_type(2))) float v2f;
typedef __attribute__((ext_vector_type(8))) float v8f;

// ---------------------------------------------------------------------------
// Stage 0: zero the BN-statistics scratch (atomically accumulated later)
// ---------------------------------------------------------------------------
__global__ void zero_k(float* __restrict__ p, int n) {
    int i = blockIdx.x * blockDim.x + threadIdx.x;
    if (i < n) p[i] = 0.0f;
}

// ---------------------------------------------------------------------------
// Stage 1: three_nn — per query point, 3 nearest of 1024 ref points (LDS-cached)
// grid = B * (N/256) = 256 blocks, block = 256 threads, one thread per point
// ---------------------------------------------------------------------------
__global__ __launch_bounds__(256) void three_nn_kernel(
    const float* __restrict__ xyz1, const float* __restrict__ xyz2,
    int* __restrict__ oidx, float* __restrict__ ow) {
    __shared__ float sxyz[3072];  // 1024 x 3 floats = 12 KB
    const int tid = threadIdx.x;
    const int b = blockIdx.x >> 4;
    const int n = ((blockIdx.x & 15) << 8) + tid;
#pragma unroll
    for (int j = 0; j < 3; ++j) {
        int q = (tid + j * 256) << 2;  // float4 granularity
        *(float4*)&sxyz[q] = *(const float4*)&xyz2[b * 3072 + q];
    }
    __syncthreads();
    const int p = (b << 12) + n;
    const float px = xyz1[p * 3 + 0];
    const float py = xyz1[p * 3 + 1];
    const float pz = xyz1[p * 3 + 2];
    float d0 = 3.4e38f, d1 = 3.4e38f, d2 = 3.4e38f;
    int i0 = 0, i1 = 0, i2 = 0;
    for (int m = 0; m < 1024; ++m) {
        float dx = px - sxyz[m * 3 + 0];
        float dy = py - sxyz[m * 3 + 1];
        float dz = pz - sxyz[m * 3 + 2];
        float d = dx * dx + dy * dy + dz * dz;
        if (d < d0)      { d2 = d1; i2 = i1; d1 = d0; i1 = i0; d0 = d; i0 = m; }
        else if (d < d1) { d2 = d1; i2 = i1; d1 = d;  i1 = m; }
        else if (d < d2) { d2 = d;  i2 = m; }
    }
    d0 = fmaxf(d0, 1e-10f); d1 = fmaxf(d1, 1e-10f); d2 = fmaxf(d2, 1e-10f);
    float w0 = 1.0f / d0, w1 = 1.0f / d1, w2 = 1.0f / d2;
    float inv = 1.0f / (w0 + w1 + w2);
    ow[p * 3 + 0] = w0 * inv;
    ow[p * 3 + 1] = w1 * inv;
    ow[p * 3 + 2] = w2 * inv;
    oidx[p * 3 + 0] = i0;
    oidx[p * 3 + 1] = i1;
    oidx[p * 3 + 2] = i2;
}

// ---------------------------------------------------------------------------
// Stage 2: weighted 3-gather interpolation + concat -> X[65536, 384]
// thread per (row, float4-chunk): 96 chunks/row (64 interp + 32 copy)
// ---------------------------------------------------------------------------
__global__ __launch_bounds__(256) void interp_concat_kernel(
    const float* __restrict__ points1, const float* __restrict__ points2,
    const int* __restrict__ idx, const float* __restrict__ w,
    float* __restrict__ X) {
    int gid = blockIdx.x * 256 + threadIdx.x;
    int row = gid / 96;
    int ch  = gid - row * 96;
    int b   = row >> 12;
    if (ch < 64) {
        int c = ch << 2;
        int base = row * 3;
        float w0 = w[base + 0], w1 = w[base + 1], w2 = w[base + 2];
        int   i0 = idx[base + 0], i1 = idx[base + 1], i2 = idx[base + 2];
        const int pb = b << 10;
        float4 p0 = *(const float4*)&points2[(pb + i0) * 256 + c];
        float4 p1 = *(const float4*)&points2[(pb + i1) * 256 + c];
        float4 p2 = *(const float4*)&points2[(pb + i2) * 256 + c];
        float4 o;
        o.x = w0 * p0.x + w1 * p1.x + w2 * p2.x;
        o.y = w0 * p0.y + w1 * p1.y + w2 * p2.y;
        o.z = w0 * p0.z + w1 * p1.z + w2 * p2.z;
        o.w = w0 * p0.w + w1 * p1.w + w2 * p2.w;
        *(float4*)&X[row * 384 + c] = o;
    } else {
        int c = (ch - 64) << 2;
        *(float4*)&X[row * 384 + 256 + c] = *(const float4*)&points1[row * 128 + c];
    }
}

// ---------------------------------------------------------------------------
// Stage 3/6: GEMM + bias via V_WMMA_F32_16X16X4_F32.
// Block = 256 threads = 8 waves; block tile 128 rows x 64 cols; each wave owns
// a 16x64 strip (4 accumulators, A-fragment reuse x4).
//  - A tile (128x32) staged in LDS, row stride 36 (8B-aligned, conflict-free
//    v2f fragment reads via ds_load).
//  - W chunk (32x64) staged in LDS in K-pair-interleaved layout
//    sW[khalf*160 + 2*n + parity], so a B fragment is ONE ds_load_b64; row
//    stride 160 => lane bank-pairs (16*khalf + n) mod 32 cover all 32 pairs
//    across both lane halves -> conflict-free, and no 8x inter-wave redundant
//    global traffic.
// ---------------------------------------------------------------------------
template <int K, int NC>
__global__ __launch_bounds__(256) void gemm_bias_wmma(
    const float* __restrict__ A, const float* __restrict__ W,
    const float* __restrict__ bias, float* __restrict__ Y) {
    __shared__ float sA[128 * 36];  // 18 KB
    __shared__ float sW[16 * 160];  // 10 KB
    const int tid  = threadIdx.x;
    const int lane = tid & 31;
    const int wave = tid >> 5;
    const int half = lane >> 4;   // 0: K={0,1}, 1: K={2,3}
    const int mrow = lane & 15;   // A row index / B,C column index
    const int row0 = blockIdx.x * 128;
    const int n0   = blockIdx.y * 64;
    const int m0   = wave * 16;

    v8f acc[4];
#pragma unroll
    for (int t = 0; t < 4; ++t)
#pragma unroll
        for (int e = 0; e < 8; ++e) acc[t][e] = 0.0f;

    for (int k0 = 0; k0 < K; k0 += 32) {
        __syncthreads();
        // Cooperative coalesced load of A[row0:row0+128, k0:k0+32] into LDS
#pragma unroll
        for (int j = 0; j < 4; ++j) {
            int i4 = tid + j * 256;       // 1024 float4 chunks
            int r  = i4 >> 3;             // 8 chunks per row
            int c  = (i4 & 7) << 2;
            *(float4*)&sA[r * 36 + c] =
                *(const float4*)&A[(row0 + r) * K + k0 + c];
        }
        // Cooperative load of W[k0:k0+32, n0:n0+64], K-pair interleaved store
#pragma unroll
        for (int j = 0; j < 2; ++j) {
            int i4 = tid + j * 256;       // 512 float4 chunks
            int k  = i4 >> 4;             // 16 chunks per K-row of 64 cols
            int c4 = (i4 & 15) << 2;
            float4 v = *(const float4*)&W[(k0 + k) * NC + n0 + c4];
            float* dst = &sW[(k >> 1) * 160 + (k & 1)];
            dst[2 * (c4 + 0)] = v.x;
            dst[2 * (c4 + 1)] = v.y;
            dst[2 * (c4 + 2)] = v.z;
            dst[2 * (c4 + 3)] = v.w;
        }
        // Pull next A K-chunk toward the WGP while we compute
        if (k0 + 32 < K) {
            __builtin_prefetch(&A[(row0 + (tid >> 1)) * K + k0 + 32 +
                                  ((tid & 1) << 4)], 0, 1);
        }
        __syncthreads();
#pragma unroll
        for (int kk = 0; kk < 32; kk += 4) {
            // A fragment: lane<16 -> K=kk..kk+1; lane>=16 -> K=kk+2..kk+3
            v2f a = *(const v2f*)&sA[(m0 + mrow) * 36 + kk + 2 * half];
            const int khalf = (kk >> 1) + half;
#pragma unroll
            for (int t = 0; t < 4; ++t) {
                int nrel = t * 16 + mrow;
                v2f bf = *(const v2f*)&sW[khalf * 160 + 2 * nrel];
                acc[t] = __builtin_amdgcn_wmma_f32_16x16x4_f32(
                    false, a, false, bf, (short)0, acc[t], false, false);
            }
        }
    }
    // Bias add + store. C layout: VGPR r -> row m0+r (lanes 0-15) / m0+r+8
    // (lanes 16-31); column = n0 + t*16 + (lane&15).
#pragma unroll
    for (int t = 0; t < 4; ++t) {
        int n = n0 + t * 16 + mrow;
        float bv = bias[n];
        int rbase = row0 + m0 + 8 * half;
#pragma unroll
        for (int r = 0; r < 8; ++r) {
            Y[(rbase + r) * NC + n] = acc[t][r] + bv;
        }
    }
}

// ---------------------------------------------------------------------------
// Stage 4/7: per-column sum & sum-of-squares (BN statistics), atomics to scratch
// ---------------------------------------------------------------------------
template <int C>
__global__ void colstat(const float* __restrict__ Y, float* __restrict__ stat,
                        int rows_per_block) {
    int c  = threadIdx.x;
    int r0 = blockIdx.x * rows_per_block;
    float s = 0.0f, s2 = 0.0f;
    for (int r = 0; r < rows_per_block; ++r) {
        float v = Y[(r0 + r) * C + c];
        s += v;
        s2 += v * v;
    }
    atomicAdd(&stat[c], s);
    atomicAdd(&stat[C + c], s2);
}

// ---------------------------------------------------------------------------
// Stage 5/8: fold mean/var + gamma/beta into per-channel scale/shift
// ---------------------------------------------------------------------------
template <int C>
__global__ void bn_params(const float* __restrict__ stat,
                          const float* __restrict__ g,
                          const float* __restrict__ beta,
                          float* __restrict__ prm, float inv_rows) {
    int c = threadIdx.x;
    float mu  = stat[c] * inv_rows;
    float var = stat[C + c] * inv_rows - mu * mu;
    float sc  = g[c] * rsqrtf(var + 1e-3f);
    prm[c]     = sc;
    prm[C + c] = beta[c] - mu * sc;
}

// ---------------------------------------------------------------------------
// Stage 5b/8b: in-place BN + ReLU, float4 vectorized
// ---------------------------------------------------------------------------
template <int C>
__global__ __launch_bounds__(256) void bn_relu(float* __restrict__ Y,
                                               const float* __restrict__ prm,
                                               int n4) {
    int i = blockIdx.x * 256 + threadIdx.x;
    if (i >= n4) return;
    int c0 = (i << 2) & (C - 1);
    float4 v = *(float4*)&Y[i << 2];
    float4 s = *(const float4*)&prm[c0];
    float4 h = *(const float4*)&prm[C + c0];
    v.x = fmaxf(v.x * s.x + h.x, 0.0f);
    v.y = fmaxf(v.y * s.y + h.y, 0.0f);
    v.z = fmaxf(v.z * s.z + h.z, 0.0f);
    v.w = fmaxf(v.w * s.w + h.w, 0.0f);
    *(float4*)&Y[i << 2] = v;
}

// ---------------------------------------------------------------------------
extern "C" void kernel_launch(void* const* d_in, const int* in_sizes, int n_in,
                              void* d_out, int out_size, void* d_ws, size_t ws_size,
                              hipStream_t stream) {
    (void)in_sizes; (void)n_in; (void)out_size; (void)ws_size;
    const float* xyz1    = (const float*)d_in[0];
    const float* points1 = (const float*)d_in[1];
    const float* xyz2    = (const float*)d_in[2];
    const float* points2 = (const float*)d_in[3];
    const float* W0      = (const float*)d_in[4];
    const float* b0      = (const float*)d_in[5];
    const float* g0      = (const float*)d_in[6];
    const float* be0     = (const float*)d_in[7];
    const float* W1      = (const float*)d_in[8];
    const float* b1      = (const float*)d_in[9];
    const float* g1      = (const float*)d_in[10];
    const float* be1     = (const float*)d_in[11];
    float* out = (float*)d_out;

    // Workspace layout (all offsets 16B aligned)
    char* ws = (char*)d_ws;
    int*   idx   = (int*)ws;                               // 196608 ints
    float* w     = (float*)(ws + 786432);                  // 196608 floats
    float* X     = (float*)(ws + 1572864);                 // 65536*384 floats (96 MB)
    float* Y0    = (float*)(ws + 1572864 + 100663296);     // 65536*256 floats (64 MB)
    float* stats = (float*)(ws + 169345024);               // 512 (L0) + 256 (L1) floats
    float* prm0  = stats + 768;                            // 512 floats
    float* prm1  = prm0 + 512;                             // 256 floats

    // 0) zero BN-stat accumulators (poisoned by harness; must re-zero each call)
    zero_k<<<1, 768, 0, stream>>>(stats, 768);

    // 1) three-nearest-neighbor search
    three_nn_kernel<<<256, 256, 0, stream>>>(xyz1, xyz2, idx, w);

    // 2) interpolation + concat -> X[65536, 384]
    interp_concat_kernel<<<24576, 256, 0, stream>>>(points1, points2, idx, w, X);

    // 3) GEMM1 + bias: Y0 = X @ W0 + b0   (fp32 WMMA)
    {
        dim3 g(512, 4);
        gemm_bias_wmma<384, 256><<<g, 256, 0, stream>>>(X, W0, b0, Y0);
    }
    // 4) BN stats, 5) fold params, 5b) BN+ReLU in place
    colstat<256><<<64, 256, 0, stream>>>(Y0, stats, 1024);
    bn_params<256><<<1, 256, 0, stream>>>(stats, g0, be0, prm0, 1.0f / 65536.0f);
    bn_relu<256><<<16384, 256, 0, stream>>>(Y0, prm0, 4194304);

    // 6) GEMM2 + bias: out = Y0 @ W1 + b1   (fp32 WMMA, straight to d_out)
    {
        dim3 g(512, 2);
        gemm_bias_wmma<256, 128><<<g, 256, 0, stream>>>(Y0, W1, b1, out);
    }
    // 7) BN stats, 8) fold params, 8b) BN+ReLU in place on d_out
    colstat<128><<<64, 128, 0, stream>>>(out, stats + 512, 1024);
    bn_params<128><<<1, 128, 0, stream>>>(stats + 512, g1, be1, prm1, 1.0f / 65536.0f);
    bn_relu<128><<<8192, 256, 0, stream>>>(out, prm1, 2097152);
}